// HamiltonianOperator_20126216749539
// MI455X (gfx1250) — compile-verified
//
#include <hip/hip_runtime.h>
#include <math.h>

// ---------------------------------------------------------------------------
// HamiltonianOperator forward for MI455X (gfx1250, wave32, WMMA).
// All matrix contractions use V_WMMA_F32_16X16X4_F32 (fp32-exact path).
// GEMM uses GLOBAL_LOAD_ASYNC_TO_LDS_B128 double buffering (ASYNCcnt) when
// the toolchain exposes the gfx1250 async builtins (guarded by __has_builtin).
// B=4, T=2048, DIM=1024, H=8, HD=128.
// ---------------------------------------------------------------------------

#define BB   4
#define TT   2048
#define DIMM 1024
#define HH   8
#define HDD  128

#if __has_builtin(__builtin_amdgcn_global_load_async_to_lds_b128) && \
    __has_builtin(__builtin_amdgcn_s_wait_asynccnt)
#define USE_ASYNC_LDS 1
#else
#define USE_ASYNC_LDS 0
#endif

typedef __attribute__((ext_vector_type(2))) float v2f;
typedef __attribute__((ext_vector_type(8))) float v8f;

__device__ __forceinline__ v8f wmma_f32(v2f a, v2f b, v8f c) {
  // D(16x16,f32) = A(16x4,f32) * B(4x16,f32) + C
  return __builtin_amdgcn_wmma_f32_16x16x4_f32(
      /*neg_a=*/false, a, /*neg_b=*/false, b,
      /*c_mod=*/(short)0, c, /*reuse_a=*/false, /*reuse_b=*/false);
}

#if USE_ASYNC_LDS
// Builtin prototype (from hipcc diagnostic): b128 takes pointers to a 16-byte
// int vector; src in AS1 (global/"__device__"), dst in AS3 (LDS).
typedef __attribute__((vector_size(16))) int ai4;
typedef __attribute__((address_space(1))) ai4* async_gptr;
typedef __attribute__((address_space(3))) ai4* async_lptr;
#define ASYNC_GL(p)  ((async_gptr)(void*)(p))
#define ASYNC_LDS(p) ((async_lptr)(void*)(p))
#endif

__device__ __forceinline__ float blockReduceSum256(float v, float* sbuf) {
  int tid = threadIdx.x;
#pragma unroll
  for (int m = 16; m >= 1; m >>= 1) v += __shfl_xor(v, m, 32);
  if ((tid & 31) == 0) sbuf[tid >> 5] = v;
  __syncthreads();
  float tot = 0.f;
#pragma unroll
  for (int i = 0; i < 8; ++i) tot += sbuf[i];
  __syncthreads();
  return tot;
}

// ---------------------------------------------------------------------------
// Kernel 1: x = LN(q + u_q); vd = sigmoid(10*local_var) * difficulty
// One 256-thread block per (b,t) row.
// ---------------------------------------------------------------------------
__global__ void pre_kernel(const float* __restrict__ state,
                           const float* __restrict__ u,
                           const float* __restrict__ g,
                           const float* __restrict__ bta,
                           float* __restrict__ x, float* __restrict__ vd) {
  __shared__ float sbuf[8];
  int idx = blockIdx.x;                  // b*T + t
  int b = idx / TT, t = idx - b * TT;
  int tp = (t == 0) ? (TT - 1) : (t - 1);
  const float* qr  = state + (size_t)idx * (2 * DIMM);
  const float* qpr = state + ((size_t)b * TT + tp) * (2 * DIMM);
  const float* ur  = u + (size_t)idx * (2 * DIMM);
  float vq[4], vu[4];
  float sx = 0, sx2 = 0, sd2 = 0, squ = 0, sqq = 0, suu = 0;
#pragma unroll
  for (int j = 0; j < 4; ++j) {
    int f = threadIdx.x + j * 256;
    float q = qr[f], qp = qpr[f], uu = ur[f];
    vq[j] = q; vu[j] = uu;
    float s = q + uu;
    sx += s; sx2 += s * s;
    float d = q - qp; sd2 += d * d;
    squ += q * uu; sqq += q * q; suu += uu * uu;
  }
  sx  = blockReduceSum256(sx, sbuf);
  sx2 = blockReduceSum256(sx2, sbuf);
  sd2 = blockReduceSum256(sd2, sbuf);
  squ = blockReduceSum256(squ, sbuf);
  sqq = blockReduceSum256(sqq, sbuf);
  suu = blockReduceSum256(suu, sbuf);
  float mu  = sx / DIMM;
  float var = sx2 / DIMM - mu * mu;
  float inv = rsqrtf(var + 1e-5f);
  if (threadIdx.x == 0) {
    float lv = sd2 / DIMM;
    float zn = fmaxf(sqrtf(sqq), 1e-8f);
    float un = fmaxf(sqrtf(suu), 1e-8f);
    float align = squ / (zn * un);
    float diff = (1.0f - align) * 0.5f;
    float vn = 1.0f / (1.0f + expf(-lv * 10.0f));
    vd[idx] = vn * diff;
  }
#pragma unroll
  for (int j = 0; j < 4; ++j) {
    int f = threadIdx.x + j * 256;
    float s = vq[j] + vu[j];
    x[(size_t)idx * DIMM + f] = (s - mu) * inv * g[f] + bta[f];
  }
}

// ---------------------------------------------------------------------------
// Kernel: h = LN(q + attn_proj)
// ---------------------------------------------------------------------------
__global__ void ln2_kernel(const float* __restrict__ state,
                           const float* __restrict__ attn,
                           const float* __restrict__ g,
                           const float* __restrict__ bta,
                           float* __restrict__ h) {
  __shared__ float sbuf[8];
  int idx = blockIdx.x;
  const float* qr = state + (size_t)idx * (2 * DIMM);
  const float* ar = attn + (size_t)idx * DIMM;
  float vv[4];
  float sx = 0, sx2 = 0;
#pragma unroll
  for (int j = 0; j < 4; ++j) {
    int f = threadIdx.x + j * 256;
    float s = qr[f] + ar[f];
    vv[j] = s; sx += s; sx2 += s * s;
  }
  sx  = blockReduceSum256(sx, sbuf);
  sx2 = blockReduceSum256(sx2, sbuf);
  float mu = sx / DIMM, var = sx2 / DIMM - mu * mu;
  float inv = rsqrtf(var + 1e-5f);
#pragma unroll
  for (int j = 0; j < 4; ++j) {
    int f = threadIdx.x + j * 256;
    h[(size_t)idx * DIMM + f] = (vv[j] - mu) * inv * g[f] + bta[f];
  }
}

// ---------------------------------------------------------------------------
// Generic fp32 WMMA GEMM: C[M,N] = act(A[M,K] @ W[N,K]^T + bias)
// 256 threads = 8 waves; 64x64 block tile; 32-deep K tiles in padded LDS.
// Async path: double-buffered GLOBAL_LOAD_ASYNC_TO_LDS_B128 (ASYNCcnt),
// one barrier per K tile.  Fallback: single-buffer VGPR-staged loads.
// ---------------------------------------------------------------------------
__global__ void __launch_bounds__(256)
gemm_kernel(const float* __restrict__ A, const float* __restrict__ W,
            const float* __restrict__ bias, float* __restrict__ C,
            int M, int N, int K, int act) {
  (void)M;
  int bc = blockIdx.x, br = blockIdx.y;
  int wave = threadIdx.x >> 5;
  int wr = wave >> 1, wc = wave & 1;      // 4x2 wave grid, 16x32 per wave
  int lane = threadIdx.x & 31;
  int half = lane >> 4, lm = lane & 15;
  int m0 = br * 64, n0 = bc * 64;
  v8f acc0 = {}, acc1 = {};

#if USE_ASYNC_LDS
  __shared__ float As[2][64][36];   // rows = 144B -> b128/float2 aligned
  __shared__ float Bs[2][64][36];
  {
    // prologue: issue async DMA for K-tile 0 into buffer 0
    for (int i = threadIdx.x; i < 512; i += 256) {
      int row = i >> 3, vv = i & 7;
      __builtin_amdgcn_global_load_async_to_lds_b128(
          ASYNC_GL(&A[(size_t)(m0 + row) * K + vv * 4]),
          ASYNC_LDS(&As[0][row][vv * 4]), 0, 0);
      __builtin_amdgcn_global_load_async_to_lds_b128(
          ASYNC_GL(&W[(size_t)(n0 + row) * K + vv * 4]),
          ASYNC_LDS(&Bs[0][row][vv * 4]), 0, 0);
    }
  }
  int cur = 0;
  for (int k0 = 0; k0 < K; k0 += 32) {
    __builtin_amdgcn_s_wait_asynccnt(0);  // own DMAs done
    __syncthreads();                       // => all waves' DMAs visible;
                                           //    prev compute on cur^1 finished
    if (k0 + 32 < K) {
      int nxt = cur ^ 1;
      for (int i = threadIdx.x; i < 512; i += 256) {
        int row = i >> 3, vv = i & 7;
        __builtin_amdgcn_global_load_async_to_lds_b128(
            ASYNC_GL(&A[(size_t)(m0 + row) * K + k0 + 32 + vv * 4]),
            ASYNC_LDS(&As[nxt][row][vv * 4]), 0, 0);
        __builtin_amdgcn_global_load_async_to_lds_b128(
            ASYNC_GL(&W[(size_t)(n0 + row) * K + k0 + 32 + vv * 4]),
            ASYNC_LDS(&Bs[nxt][row][vv * 4]), 0, 0);
      }
    }
#pragma unroll
    for (int kk = 0; kk < 8; ++kk) {
      v2f a  = *(const v2f*)&As[cur][wr * 16 + lm][kk * 4 + 2 * half];
      v2f b0 = *(const v2f*)&Bs[cur][wc * 32 + lm][kk * 4 + 2 * half];
      v2f b1 = *(const v2f*)&Bs[cur][wc * 32 + 16 + lm][kk * 4 + 2 * half];
      acc0 = wmma_f32(a, b0, acc0);
      acc1 = wmma_f32(a, b1, acc1);
    }
    cur ^= 1;
  }
#else
  __shared__ float As[64][36];   // rows = 144B -> float4/float2 aligned
  __shared__ float Bs[64][36];   // Bs[n][k] = W[n0+n][k0+k] = B-matrix(k,n)
  for (int k0 = 0; k0 < K; k0 += 32) {
    __syncthreads();
    for (int i = threadIdx.x; i < 512; i += 256) {
      int row = i >> 3, vv = i & 7;
      *(float4*)&As[row][vv * 4] =
          *(const float4*)&A[(size_t)(m0 + row) * K + k0 + vv * 4];
      *(float4*)&Bs[row][vv * 4] =
          *(const float4*)&W[(size_t)(n0 + row) * K + k0 + vv * 4];
    }
    if (k0 + 32 < K) {  // CDNA5 global_prefetch_b8: warm next K tile
      __builtin_prefetch(&A[(size_t)(m0 + (threadIdx.x >> 2)) * K + k0 + 32], 0, 0);
      __builtin_prefetch(&W[(size_t)(n0 + (threadIdx.x >> 2)) * K + k0 + 32], 0, 0);
    }
    __syncthreads();
#pragma unroll
    for (int kk = 0; kk < 8; ++kk) {
      v2f a  = *(const v2f*)&As[wr * 16 + lm][kk * 4 + 2 * half];
      v2f b0 = *(const v2f*)&Bs[wc * 32 + lm][kk * 4 + 2 * half];
      v2f b1 = *(const v2f*)&Bs[wc * 32 + 16 + lm][kk * 4 + 2 * half];
      acc0 = wmma_f32(a, b0, acc0);
      acc1 = wmma_f32(a, b1, acc1);
    }
  }
#endif

  int rowB = m0 + wr * 16 + 8 * half;     // C layout: vgpr r -> M = r + 8*half
  int colB = n0 + wc * 32;
#pragma unroll
  for (int r = 0; r < 8; ++r) {
    int row = rowB + r;
    int c0 = colB + lm, c1 = colB + 16 + lm;
    float v0 = acc0[r] + (bias ? bias[c0] : 0.0f);
    float v1 = acc1[r] + (bias ? bias[c1] : 0.0f);
    if (act) {  // exact GELU
      v0 = 0.5f * v0 * (1.0f + erff(v0 * 0.70710678118654752f));
      v1 = 0.5f * v1 * (1.0f + erff(v1 * 0.70710678118654752f));
    }
    C[(size_t)row * N + c0] = v0;
    C[(size_t)row * N + c1] = v1;
  }
}

// ---------------------------------------------------------------------------
// Flash attention: one wave32 per (16 query rows, head, batch).
// Q frags resident in registers; K/V fragments gathered from L2-resident qkv;
// P re-shaped C-layout -> A-layout through a 1KB wave-private LDS tile.
// ---------------------------------------------------------------------------
__global__ void __launch_bounds__(32)
attn_kernel(const float* __restrict__ qkv, float* __restrict__ attn_raw) {
  __shared__ float Plds[16][18];
  int qt = blockIdx.x, h = blockIdx.y, b = blockIdx.z;
  int lane = threadIdx.x & 31;
  int half = lane >> 4, lm = lane & 15;
  int q0 = qt * 16;
  const float scale = 0.08838834764831845f;  // 1/sqrt(128), folded into Q
  const float* qrow = qkv + ((size_t)(b * TT + q0 + lm)) * (3 * DIMM) + h * HDD;
  v2f a[32];
#pragma unroll
  for (int ks = 0; ks < 32; ++ks) {
    v2f t = *(const v2f*)(qrow + ks * 4 + 2 * half);
    a[ks][0] = t[0] * scale;
    a[ks][1] = t[1] * scale;
  }
  v8f o[8];
  v8f vzero = {};
#pragma unroll
  for (int c = 0; c < 8; ++c) o[c] = vzero;
  float mrow[8], lrow[8];
#pragma unroll
  for (int r = 0; r < 8; ++r) { mrow[r] = -3.0e38f; lrow[r] = 0.f; }

  for (int kb = 0; kb < TT / 16; ++kb) {
    int kbase = kb * 16;
    // S = Q @ K^T  (B(k,n) = K[key n][feat k], per-lane float2 gather)
    const float* krow =
        qkv + ((size_t)(b * TT + kbase + lm)) * (3 * DIMM) + DIMM + h * HDD;
    v8f s = {};
#pragma unroll
    for (int ks = 0; ks < 32; ++ks) {
      v2f bk = *(const v2f*)(krow + ks * 4 + 2 * half);
      s = wmma_f32(a[ks], bk, s);
    }
    // online softmax on C-layout rows (row r+8*half lives in 16-lane groups)
    float scl[8];
#pragma unroll
    for (int r = 0; r < 8; ++r) {
      float sv = s[r];
      float rmax = sv;
#pragma unroll
      for (int m = 8; m >= 1; m >>= 1) rmax = fmaxf(rmax, __shfl_xor(rmax, m, 16));
      float newm = fmaxf(mrow[r], rmax);
      float p = expf(sv - newm);
      float psum = p;
#pragma unroll
      for (int m = 8; m >= 1; m >>= 1) psum += __shfl_xor(psum, m, 16);
      float sc = expf(mrow[r] - newm);
      lrow[r] = lrow[r] * sc + psum;
      mrow[r] = newm;
      scl[r] = sc;
      Plds[r + 8 * half][lm] = p;  // C-layout -> LDS (row-major 16x16)
    }
#pragma unroll
    for (int c = 0; c < 8; ++c)
#pragma unroll
      for (int r = 0; r < 8; ++r) o[c][r] *= scl[r];
    // O += P @ V   (A = P from LDS in A-layout; B(k,n) = V[key k][col n])
    const float* vbase =
        qkv + ((size_t)(b * TT + kbase)) * (3 * DIMM) + 2 * DIMM + h * HDD;
#pragma unroll
    for (int ks2 = 0; ks2 < 4; ++ks2) {
      v2f ap = *(const v2f*)&Plds[lm][ks2 * 4 + 2 * half];
      int kv = ks2 * 4 + 2 * half;
#pragma unroll
      for (int c = 0; c < 8; ++c) {
        int n = c * 16 + lm;
        v2f bv;
        bv[0] = vbase[(size_t)kv * (3 * DIMM) + n];
        bv[1] = vbase[(size_t)(kv + 1) * (3 * DIMM) + n];
        o[c] = wmma_f32(ap, bv, o[c]);
      }
    }
  }
  float inv_l[8];
#pragma unroll
  for (int r = 0; r < 8; ++r) inv_l[r] = 1.0f / lrow[r];
#pragma unroll
  for (int c = 0; c < 8; ++c) {
#pragma unroll
    for (int r = 0; r < 8; ++r) {
      int row = q0 + r + 8 * half;
      int col = h * HDD + c * 16 + lm;
      attn_raw[((size_t)(b * TT + row)) * DIMM + col] = o[c][r] * inv_l[r];
    }
  }
}

// ---------------------------------------------------------------------------
// Streaming softmax-entropy of S = q q^T / sqrt(DIM) per row:
//   entropy = logZ - E[s];  track (max, Z'=sum e^{s-m}, Y=sum s e^{s-m}).
// One wave32 per 16 query rows; fragments gathered from L2-resident state.
// ---------------------------------------------------------------------------
__global__ void __launch_bounds__(32)
entropy_kernel(const float* __restrict__ state, float* __restrict__ ent_row) {
  int qt = blockIdx.x, b = blockIdx.y;
  int lane = threadIdx.x & 31;
  int half = lane >> 4, lm = lane & 15;
  int q0 = qt * 16;
  const float* qrow = state + ((size_t)(b * TT + q0 + lm)) * (2 * DIMM);
  float mrow[8], Z[8], Y[8];
#pragma unroll
  for (int r = 0; r < 8; ++r) { mrow[r] = -3.0e38f; Z[r] = 0.f; Y[r] = 0.f; }
  const float isq = 0.03125f;  // 1/sqrt(1024)
  for (int kb = 0; kb < TT / 16; ++kb) {
    const float* krow = state + ((size_t)(b * TT + kb * 16 + lm)) * (2 * DIMM);
    v8f s = {};
#pragma unroll 4
    for (int ks = 0; ks < 256; ++ks) {
      v2f aq = *(const v2f*)(qrow + ks * 4 + 2 * half);
      v2f bq = *(const v2f*)(krow + ks * 4 + 2 * half);
      s = wmma_f32(aq, bq, s);
    }
#pragma unroll
    for (int r = 0; r < 8; ++r) {
      float sv = s[r] * isq;
      float rmax = sv;
#pragma unroll
      for (int m = 8; m >= 1; m >>= 1) rmax = fmaxf(rmax, __shfl_xor(rmax, m, 16));
      float newm = fmaxf(mrow[r], rmax);
      float e = expf(sv - newm);
      float esum = e, ysum = sv * e;
#pragma unroll
      for (int m = 8; m >= 1; m >>= 1) {
        esum += __shfl_xor(esum, m, 16);
        ysum += __shfl_xor(ysum, m, 16);
      }
      float sc = expf(mrow[r] - newm);
      Z[r] = Z[r] * sc + esum;
      Y[r] = Y[r] * sc + ysum;
      mrow[r] = newm;
    }
  }
#pragma unroll
  for (int r = 0; r < 8; ++r) {
    if (lm == 0) {
      float ent = (mrow[r] + logf(Z[r])) - Y[r] / Z[r];
      ent_row[(size_t)b * TT + q0 + r + 8 * half] = ent;
    }
  }
}

__global__ void entropy_reduce(const float* __restrict__ ent_row,
                               float* __restrict__ ent_norm) {
  __shared__ float sbuf[8];
  int b = blockIdx.x;
  float s = 0.f;
  for (int t = threadIdx.x; t < TT; t += 256) s += ent_row[(size_t)b * TT + t];
  s = blockReduceSum256(s, sbuf);
  if (threadIdx.x == 0) ent_norm[b] = (s / (float)TT) / 10.0f;
}

__global__ void dt_kernel(const float* __restrict__ vd,
                          const float* __restrict__ en,
                          float* __restrict__ dt) {
  int i = blockIdx.x * 256 + threadIdx.x;
  if (i >= BB * TT) return;
  float c = vd[i] * en[i / TT];
  c = fminf(fmaxf(c, 0.f), 1.f);
  dt[i] = 0.05f + 0.25f * (1.f - c);   // DT_MIN + (DT_MAX-DT_MIN)*(1-c)
}

__global__ void pnew_kernel(const float* __restrict__ state,
                            const float* __restrict__ u,
                            const float* __restrict__ force,
                            const float* __restrict__ dt,
                            float* __restrict__ out, float* __restrict__ pnu) {
  size_t i = (size_t)blockIdx.x * 256 + threadIdx.x;
  if (i >= (size_t)BB * TT * DIMM) return;
  size_t row = i / DIMM;
  int d = (int)(i - row * DIMM);
  float p = state[row * (2 * DIMM) + DIMM + d];
  float pn = p + dt[row] * (force[i] - 0.05f * p);   // GAMMA = 0.05
  out[row * (2 * DIMM) + DIMM + d] = pn;
  pnu[i] = pn + u[row * (2 * DIMM) + DIMM + d];
}

__global__ void qnew_kernel(const float* __restrict__ state,
                            const float* __restrict__ dt,
                            const float* __restrict__ vel,
                            float* __restrict__ out) {
  size_t i = (size_t)blockIdx.x * 256 + threadIdx.x;
  if (i >= (size_t)BB * TT * DIMM) return;
  size_t row = i / DIMM;
  int d = (int)(i - row * DIMM);
  out[row * (2 * DIMM) + d] = state[row * (2 * DIMM) + d] + dt[row] * vel[i];
}

// ---------------------------------------------------------------------------
// Launcher. Workspace regions are aliased across pipeline phases:
//   x(32MB: x -> h -> velocity) | big(128MB: qkv -> ff_mid)
//   araw(32MB: attn_raw -> pnu) | aproj(32MB: attn_proj -> force) | small stats
// Total ~235MB of d_ws.
// ---------------------------------------------------------------------------
extern "C" void kernel_launch(void* const* d_in, const int* in_sizes, int n_in,
                              void* d_out, int out_size, void* d_ws,
                              size_t ws_size, hipStream_t stream) {
  (void)in_sizes; (void)n_in; (void)out_size; (void)ws_size;
  const float* state = (const float*)d_in[0];
  const float* u     = (const float*)d_in[1];
  const float* massw = (const float*)d_in[2];
  const float* ln1g  = (const float*)d_in[3];
  const float* ln1b  = (const float*)d_in[4];
  const float* inw   = (const float*)d_in[5];
  const float* inb   = (const float*)d_in[6];
  const float* outw  = (const float*)d_in[7];
  const float* outb  = (const float*)d_in[8];
  const float* ln2g  = (const float*)d_in[9];
  const float* ln2b  = (const float*)d_in[10];
  const float* ffw1  = (const float*)d_in[11];
  const float* ffb1  = (const float*)d_in[12];
  const float* ffw2  = (const float*)d_in[13];
  const float* ffb2  = (const float*)d_in[14];
  float* out = (float*)d_out;
  float* ws  = (float*)d_ws;

  const size_t M = (size_t)BB * TT;  // 8192 rows
  float* x     = ws;                  // M*DIMM   : x -> h -> velocity
  float* big   = x + M * DIMM;        // M*4*DIMM : qkv -> ff_mid
  float* araw  = big + M * 4 * DIMM;  // M*DIMM   : attn_raw -> pnu
  float* aproj = araw + M * DIMM;     // M*DIMM   : attn_proj -> force
  float* vd    = aproj + M * DIMM;    // M
  float* erow  = vd + M;              // M
  float* enorm = erow + M;            // BB
  float* dtb   = enorm + 16;          // M

  // 1) x = LN(q+u_q), per-row variance_norm*difficulty
  pre_kernel<<<dim3((unsigned)M), 256, 0, stream>>>(state, u, ln1g, ln1b, x, vd);
  // 2) qkv = x @ in_proj_w^T + b
  gemm_kernel<<<dim3(3 * DIMM / 64, (unsigned)(M / 64)), 256, 0, stream>>>(
      x, inw, inb, big, (int)M, 3 * DIMM, DIMM, 0);
  // 3) flash attention
  attn_kernel<<<dim3(TT / 16, HH, BB), 32, 0, stream>>>(big, araw);
  // 4) out projection
  gemm_kernel<<<dim3(DIMM / 64, (unsigned)(M / 64)), 256, 0, stream>>>(
      araw, outw, outb, aproj, (int)M, DIMM, DIMM, 0);
  // 5) h = LN(q + attn)
  ln2_kernel<<<dim3((unsigned)M), 256, 0, stream>>>(state, aproj, ln2g, ln2b, x);
  // 6) ff_mid = gelu(h @ ff_w1^T + b1)
  gemm_kernel<<<dim3(4 * DIMM / 64, (unsigned)(M / 64)), 256, 0, stream>>>(
      x, ffw1, ffb1, big, (int)M, 4 * DIMM, DIMM, 1);
  // 7) force = ff_mid @ ff_w2^T + b2
  gemm_kernel<<<dim3(DIMM / 64, (unsigned)(M / 64)), 256, 0, stream>>>(
      big, ffw2, ffb2, aproj, (int)M, DIMM, 4 * DIMM, 0);
  // 8-10) semantic-metric entropy -> per-batch norm -> dt
  entropy_kernel<<<dim3(TT / 16, BB), 32, 0, stream>>>(state, erow);
  entropy_reduce<<<dim3(BB), 256, 0, stream>>>(erow, enorm);
  dt_kernel<<<dim3((BB * TT + 255) / 256), 256, 0, stream>>>(vd, enorm, dtb);
  // 11) p_new = p + dt*(force - gamma*p); pnu = p_new + u_p
  pnew_kernel<<<dim3((unsigned)((M * DIMM + 255) / 256)), 256, 0, stream>>>(
      state, u, aproj, dtb, out, araw);
  // 12) velocity = pnu @ mass_w^T
  gemm_kernel<<<dim3(DIMM / 64, (unsigned)(M / 64)), 256, 0, stream>>>(
      araw, massw, nullptr, x, (int)M, DIMM, DIMM, 0);
  // 13) q_new = q + dt*velocity
  qnew_kernel<<<dim3((unsigned)((M * DIMM + 255) / 256)), 256, 0, stream>>>(
      state, dtb, x, out);
}